// Seq2Seq_69269232550543
// MI455X (gfx1250) — compile-verified
//
#include <hip/hip_runtime.h>
#include <math.h>

// ---------------- problem dimensions (compile-time) ----------------
#define Vz 32000
#define Ez 300
#define Hz 300
#define Dz 300
#define Bz 64
#define Sz 128
#define Tz 32
#define G3 900            // 3*H = 3*D
#define C2 600            // 2*H
#define KCAT 1200         // 2H + D + E

typedef __attribute__((ext_vector_type(2))) float v2f;
typedef __attribute__((ext_vector_type(8))) float v8f;

#define LDV2(p) (*(const v2f*)(p))
#define WMMA4(a, b, c) __builtin_amdgcn_wmma_f32_16x16x4_f32(false, (a), false, (b), \
                                                             (short)0, (c), false, false)

// =====================================================================
// Generic NT GEMM:  C[m,n] = act( sum_k A[m,k] * W[n,k] + bias[n] )
//   A: M x K row-major, leading dim lda
//   W: N rows, row n at W + n*ldw (8B-aligned rows), K columns
// V_WMMA_F32_16X16X4_F32; block = 4 waves; wave w owns rows
// m0 = blockIdx.y*64 + w*16 and TWO column tiles n0 = blockIdx.x*32 (+16),
// sharing one A fragment (halves A traffic, 4 independent WMMA chains).
// Out-of-range columns: W row pointer CLAMPED to row N-1 (branch-free
// k-loop); garbage flows only into D columns that are never stored.
// k-loop is 2-stage software-pipelined: group k+8 loads issue before
// group k's WMMAs, hiding load latency inside a single wave.
// =====================================================================
__global__ __launch_bounds__(128)
void gemm_nt_f32_wmma(const float* __restrict__ A, int lda,
                      const float* __restrict__ W, int ldw,
                      const float* __restrict__ bias,
                      float* __restrict__ C, long ldc,
                      int M, int N, int K, int act)
{
    const int lane = threadIdx.x & 31;
    const int wave = threadIdx.x >> 5;
    const int n0 = blockIdx.x * 32;
    const int m0 = blockIdx.y * 64 + wave * 16;
    if (m0 >= M) return;                 // wave-uniform (EXEC stays full at WMMA)

    const int half = lane >> 4;          // 0: lanes 0-15, 1: lanes 16-31
    const int l15  = lane & 15;
    const int coff = half * 2;           // this lane supplies K = coff, coff+1
    const int mrow = m0 + l15;
    const int nc0  = n0 + l15;           // tile-0 column for this lane
    const int nc1  = nc0 + 16;           // tile-1 column for this lane
    const int nc0c = (nc0 < N) ? nc0 : (N - 1);   // clamped (always-valid) rows
    const int nc1c = (nc1 < N) ? nc1 : (N - 1);

    const float* Arow  = A + (size_t)mrow * lda + coff;
    const float* Wrow0 = W + (size_t)nc0c * ldw + coff;
    const float* Wrow1 = W + (size_t)nc1c * ldw + coff;

    v8f acc00 = {}, acc01 = {}, acc10 = {}, acc11 = {};

    const int K8 = K & ~7;
    if (K8 > 0) {
        // ---- pipeline prologue: preload group 0 ----
        v2f a0  = LDV2(Arow);        v2f a1  = LDV2(Arow + 4);
        v2f b00 = LDV2(Wrow0);       v2f b01 = LDV2(Wrow0 + 4);
        v2f b10 = LDV2(Wrow1);       v2f b11 = LDV2(Wrow1 + 4);
        // ---- steady state: issue group kk loads, compute group kk-8 ----
        for (int kk = 8; kk < K8; kk += 8) {
            v2f na0  = LDV2(Arow + kk);      v2f na1  = LDV2(Arow + kk + 4);
            v2f nb00 = LDV2(Wrow0 + kk);     v2f nb01 = LDV2(Wrow0 + kk + 4);
            v2f nb10 = LDV2(Wrow1 + kk);     v2f nb11 = LDV2(Wrow1 + kk + 4);
            __builtin_prefetch(Wrow0 + kk + 64, 0, 1);   // global_prefetch_b8
            __builtin_prefetch(Wrow1 + kk + 64, 0, 1);
            acc00 = WMMA4(a0, b00, acc00);
            acc10 = WMMA4(a0, b10, acc10);
            acc01 = WMMA4(a1, b01, acc01);
            acc11 = WMMA4(a1, b11, acc11);
            a0 = na0; a1 = na1;
            b00 = nb00; b01 = nb01; b10 = nb10; b11 = nb11;
        }
        // ---- epilogue: last full group ----
        acc00 = WMMA4(a0, b00, acc00);
        acc10 = WMMA4(a0, b10, acc10);
        acc01 = WMMA4(a1, b01, acc01);
        acc11 = WMMA4(a1, b11, acc11);
    }
    if (K & 4) {                       // K % 8 == 4 tail (K=300, 900)
        v2f ta  = LDV2(Arow + K8);
        v2f tb0 = LDV2(Wrow0 + K8);
        v2f tb1 = LDV2(Wrow1 + K8);
        acc00 = WMMA4(ta, tb0, acc00);
        acc10 = WMMA4(ta, tb1, acc10);
    }
    v8f t0 = acc00 + acc01;
    v8f t1 = acc10 + acc11;

    // ---- store (per-lane column guard, bias + optional tanh) ----
    if (nc0 < N) {
        const float bv = (bias != nullptr) ? bias[nc0] : 0.0f;
#pragma unroll
        for (int r = 0; r < 8; ++r) {
            float vv = t0[r] + bv;
            if (act == 1) vv = tanhf(vv);
            C[(size_t)(m0 + r + half * 8) * ldc + nc0] = vv;
        }
    }
    if (nc1 < N) {
        const float bv = (bias != nullptr) ? bias[nc1] : 0.0f;
#pragma unroll
        for (int r = 0; r < 8; ++r) {
            float vv = t1[r] + bv;
            if (act == 1) vv = tanhf(vv);
            C[(size_t)(m0 + r + half * 8) * ldc + nc1] = vv;
        }
    }
}

// ---------------- embedding: xs[(s*B+b)*E+e] = emb[sen]+tt_emb[tt] ----------------
__global__ void k_embed_add(const int* __restrict__ sen, const int* __restrict__ tt,
                            const float* __restrict__ emb, const float* __restrict__ tte,
                            float* __restrict__ xs)
{
    int sb = blockIdx.x;             // sb = s*B + b
    int s = sb / Bz, b = sb % Bz;
    const float* er = emb + (size_t)sen[b * Sz + s] * Ez;
    const float* tr = tte + (size_t)tt[b * Sz + s] * Ez;
    float* xr = xs + (size_t)sb * Ez;
    for (int e = threadIdx.x; e < Ez; e += blockDim.x) xr[e] = er[e] + tr[e];
}

// ---------------- decoder input embeddings for all steps ----------------
__global__ void k_embed_q(const int* __restrict__ q, const float* __restrict__ emb,
                          float* __restrict__ de)
{
    int tb = blockIdx.x;             // tb = t*B + b, t in [0, T-1)
    int t = tb / Bz, b = tb % Bz;
    const float* er = emb + (size_t)q[b * Tz + t] * Ez;
    float* xr = de + (size_t)tb * Ez;
    for (int e = threadIdx.x; e < Ez; e += blockDim.x) xr[e] = er[e];
}

// ---------------- GRU gate fusion ----------------
__global__ void k_gru_fuse(const float* __restrict__ GI, const float* __restrict__ GH,
                           float* __restrict__ h, float* __restrict__ ctx_out,
                           long ctx_bstride)
{
    int i = blockIdx.x * blockDim.x + threadIdx.x;
    if (i >= Bz * Hz) return;
    int b = i / Hz, j = i % Hz;
    const float* gi = GI + (size_t)b * G3;
    const float* gh = GH + (size_t)b * G3;
    float r = 1.0f / (1.0f + __expf(-(gi[j] + gh[j])));
    float z = 1.0f / (1.0f + __expf(-(gi[Hz + j] + gh[Hz + j])));
    float n = tanhf(gi[2 * Hz + j] + r * gh[2 * Hz + j]);
    float h2 = (1.0f - z) * n + z * h[i];
    h[i] = h2;
    if (ctx_out) ctx_out[(size_t)b * ctx_bstride + j] = h2;
}

// ---------------- attention scores: v . tanh(Ah[b]+Ac[b,s]) with PAD mask --------
__global__ __launch_bounds__(64)
void k_attn_score(const float* __restrict__ Ah, const float* __restrict__ Ac,
                  const float* __restrict__ vvec, const int* __restrict__ sen,
                  float* __restrict__ scores)
{
    int s = blockIdx.x, b = blockIdx.y;
    const float* ah = Ah + (size_t)b * Dz;
    const float* ac = Ac + ((size_t)b * Sz + s) * Dz;
    float p = 0.0f;
    for (int n = threadIdx.x; n < Dz; n += 64) p += vvec[n] * tanhf(ah[n] + ac[n]);
    __shared__ float red[64];
    red[threadIdx.x] = p; __syncthreads();
    for (int off = 32; off > 0; off >>= 1) {
        if (threadIdx.x < off) red[threadIdx.x] += red[threadIdx.x + off];
        __syncthreads();
    }
    if (threadIdx.x == 0) {
        float sc = red[0];
        scores[b * Sz + s] = (sen[b * Sz + s] != 0) ? sc : -INFINITY;
    }
}

// ---------------- softmax over S + context reduction -----------------------------
__global__ __launch_bounds__(128)
void k_softmax_ctx(const float* __restrict__ scores, const float* __restrict__ context,
                   float* __restrict__ a_out, float* __restrict__ ctx)
{
    int b = blockIdx.x, tid = threadIdx.x;
    __shared__ float red[128];
    __shared__ float sm[128];
    float e = scores[b * Sz + tid];
    red[tid] = e; __syncthreads();
    for (int off = 64; off > 0; off >>= 1) {
        if (tid < off) red[tid] = fmaxf(red[tid], red[tid + off]);
        __syncthreads();
    }
    float mx = red[0]; __syncthreads();
    float ex = __expf(e - mx);
    red[tid] = ex; __syncthreads();
    for (int off = 64; off > 0; off >>= 1) {
        if (tid < off) red[tid] += red[tid + off];
        __syncthreads();
    }
    float a = ex / red[0];
    sm[tid] = a;
    a_out[b * Sz + tid] = a;
    __syncthreads();
    const float* cb = context + (size_t)b * Sz * C2;
    for (int h = tid; h < C2; h += 128) {
        float acc = 0.0f;
#pragma unroll 4
        for (int s2 = 0; s2 < Sz; ++s2) acc += sm[s2] * cb[(size_t)s2 * C2 + h];
        ctx[(size_t)b * C2 + h] = acc;
    }
}

// ---------------- per-row column copy (for concats) ------------------------------
__global__ void k_copy_cols(const float* __restrict__ src, long sld,
                            float* __restrict__ dst, long dld, int cols)
{
    int r = blockIdx.x;
    const float* s = src + (size_t)r * sld;
    float* d = dst + (size_t)r * dld;
    for (int c = threadIdx.x; c < cols; c += blockDim.x) d[c] = s[c];
}

// ---------------- output init: zero outputs[0]/attentions[0], write generated ----
__global__ void k_init_out(float* __restrict__ out, const int* __restrict__ q)
{
    size_t i = (size_t)blockIdx.x * blockDim.x + threadIdx.x;
    if (i < (size_t)Bz * Vz) out[i] = 0.0f;
    const size_t attBase = (size_t)Tz * Bz * Vz;
    if (i < (size_t)Bz * Sz) out[attBase + i] = 0.0f;
    const size_t genBase = attBase + (size_t)Tz * Bz * Sz;
    if (i < (size_t)Bz * Tz) out[genBase + i] = (float)q[i];
}

__global__ void k_zero(float* __restrict__ p, int n)
{
    int i = blockIdx.x * blockDim.x + threadIdx.x;
    if (i < n) p[i] = 0.0f;
}

// =====================================================================
extern "C" void kernel_launch(void* const* d_in, const int* in_sizes, int n_in,
                              void* d_out, int out_size, void* d_ws, size_t ws_size,
                              hipStream_t stream)
{
    (void)in_sizes; (void)n_in; (void)out_size; (void)ws_size;
    const int*   sen     = (const int*)d_in[0];
    const int*   tt_ids  = (const int*)d_in[1];
    const int*   quest   = (const int*)d_in[2];
    const float* emb     = (const float*)d_in[3];
    const float* tte     = (const float*)d_in[4];
    const float* Wih_f   = (const float*)d_in[5];
    const float* Whh_f   = (const float*)d_in[6];
    const float* bih_f   = (const float*)d_in[7];
    const float* bhh_f   = (const float*)d_in[8];
    const float* Wih_b   = (const float*)d_in[9];
    const float* Whh_b   = (const float*)d_in[10];
    const float* bih_b   = (const float*)d_in[11];
    const float* bhh_b   = (const float*)d_in[12];
    const float* fc_W    = (const float*)d_in[13];
    const float* fc_b    = (const float*)d_in[14];
    const float* attn_W  = (const float*)d_in[15];
    const float* attn_b  = (const float*)d_in[16];
    const float* vvec    = (const float*)d_in[17];
    const float* dWih    = (const float*)d_in[18];
    const float* dWhh    = (const float*)d_in[19];
    const float* dbih    = (const float*)d_in[20];
    const float* dbhh    = (const float*)d_in[21];
    const float* out_W   = (const float*)d_in[22];
    const float* out_b   = (const float*)d_in[23];
    float* out = (float*)d_out;

    // ---- workspace carve-up (floats) ----
    float* ws = (float*)d_ws;
    size_t off = 0;
    auto alloc = [&](size_t n) { float* p = ws + off; off += n; return p; };
    float* xs      = alloc((size_t)Sz * Bz * Ez);     // time-major embedded input
    float* GI_f    = alloc((size_t)Sz * Bz * G3);     // batched input gates, fwd
    float* GI_b    = alloc((size_t)Sz * Bz * G3);     // batched input gates, bwd
    float* context = alloc((size_t)Bz * Sz * C2);     // (B,S,2H)
    float* Ac      = alloc((size_t)Bz * Sz * Dz);     // hoisted attention energies
    float* dec_e   = alloc((size_t)(Tz - 1) * Bz * Ez);
    float* h_f     = alloc((size_t)Bz * Hz);
    float* h_b     = alloc((size_t)Bz * Hz);
    float* hcat    = alloc((size_t)Bz * C2);
    float* hid     = alloc((size_t)Bz * Dz);
    float* GH      = alloc((size_t)Bz * G3);
    float* Ah      = alloc((size_t)Bz * Dz);
    float* scores  = alloc((size_t)Bz * Sz);
    float* ctx     = alloc((size_t)Bz * C2);
    float* g_in    = alloc((size_t)Bz * G3);
    float* GId     = alloc((size_t)Bz * G3);
    float* GHd     = alloc((size_t)Bz * G3);
    float* cat     = alloc((size_t)Bz * KCAT);

    const size_t attBase = (size_t)Tz * Bz * Vz;

    // ---- init outputs / h0 ----
    k_init_out<<<(Bz * Vz + 255) / 256, 256, 0, stream>>>(out, quest);
    k_zero<<<(Bz * Hz + 255) / 256, 256, 0, stream>>>(h_f, Bz * Hz);
    k_zero<<<(Bz * Hz + 255) / 256, 256, 0, stream>>>(h_b, Bz * Hz);

    // ---- embeddings ----
    k_embed_add<<<Sz * Bz, 128, 0, stream>>>(sen, tt_ids, emb, tte, xs);
    k_embed_q<<<(Tz - 1) * Bz, 128, 0, stream>>>(quest, emb, dec_e);

    // ---- batched encoder input gates: (8192x300)@(300x900)^T ----
    {
        dim3 g((G3 + 31) / 32, (Sz * Bz) / 64);
        gemm_nt_f32_wmma<<<g, 128, 0, stream>>>(xs, Ez, Wih_f, Ez, bih_f, GI_f, G3,
                                                Sz * Bz, G3, Ez, 0);
        gemm_nt_f32_wmma<<<g, 128, 0, stream>>>(xs, Ez, Wih_b, Ez, bih_b, GI_b, G3,
                                                Sz * Bz, G3, Ez, 0);
    }

    // ---- encoder recurrences ----
    dim3 ggate((G3 + 31) / 32, 1);
    for (int t = 0; t < Sz; ++t) {
        gemm_nt_f32_wmma<<<ggate, 128, 0, stream>>>(h_f, Hz, Whh_f, Hz, bhh_f, GH, G3,
                                                    Bz, G3, Hz, 0);
        k_gru_fuse<<<(Bz * Hz + 255) / 256, 256, 0, stream>>>(
            GI_f + (size_t)t * Bz * G3, GH, h_f,
            context + (size_t)t * C2, (long)Sz * C2);
    }
    for (int t = 0; t < Sz; ++t) {
        int s = Sz - 1 - t;
        gemm_nt_f32_wmma<<<ggate, 128, 0, stream>>>(h_b, Hz, Whh_b, Hz, bhh_b, GH, G3,
                                                    Bz, G3, Hz, 0);
        k_gru_fuse<<<(Bz * Hz + 255) / 256, 256, 0, stream>>>(
            GI_b + (size_t)s * Bz * G3, GH, h_b,
            context + (size_t)s * C2 + Hz, (long)Sz * C2);
    }

    // ---- decoder initial hidden: tanh([h_f,h_b] @ fc_W^T + fc_b) ----
    k_copy_cols<<<Bz, 128, 0, stream>>>(h_f, Hz, hcat, C2, Hz);
    k_copy_cols<<<Bz, 128, 0, stream>>>(h_b, Hz, hcat + Hz, C2, Hz);
    gemm_nt_f32_wmma<<<dim3((Dz + 31) / 32, 1), 128, 0, stream>>>(
        hcat, C2, fc_W, C2, fc_b, hid, Dz, Bz, Dz, C2, 1 /*tanh*/);

    // ---- hoisted attention energies: Ac = context @ attn_W[:,D:]^T + attn_b ----
    gemm_nt_f32_wmma<<<dim3((Dz + 31) / 32, (Bz * Sz) / 64), 128, 0, stream>>>(
        context, C2, attn_W + Dz, G3, attn_b, Ac, Dz, Bz * Sz, Dz, C2, 0);

    // ---- decoder loop ----
    for (int t = 0; t < Tz - 1; ++t) {
        const float* e_t = dec_e + (size_t)t * Bz * Ez;
        // Ah = hid @ attn_W[:, :D]^T   (bias folded into Ac)
        gemm_nt_f32_wmma<<<dim3((Dz + 31) / 32, 1), 128, 0, stream>>>(
            hid, Dz, attn_W, G3, nullptr, Ah, Dz, Bz, Dz, Dz, 0);
        k_attn_score<<<dim3(Sz, Bz), 64, 0, stream>>>(Ah, Ac, vvec, sen, scores);
        k_softmax_ctx<<<Bz, 128, 0, stream>>>(
            scores, context, out + attBase + (size_t)(t + 1) * Bz * Sz, ctx);

        // g_in = [e_t, ctx]
        k_copy_cols<<<Bz, 128, 0, stream>>>(e_t, Ez, g_in, G3, Ez);
        k_copy_cols<<<Bz, 128, 0, stream>>>(ctx, C2, g_in + Ez, G3, C2);

        gemm_nt_f32_wmma<<<ggate, 128, 0, stream>>>(g_in, G3, dWih, G3, dbih, GId, G3,
                                                    Bz, G3, G3, 0);
        gemm_nt_f32_wmma<<<ggate, 128, 0, stream>>>(hid, Dz, dWhh, Dz, dbhh, GHd, G3,
                                                    Bz, G3, Dz, 0);
        k_gru_fuse<<<(Bz * Hz + 255) / 256, 256, 0, stream>>>(GId, GHd, hid, nullptr, 0);

        // cat = [hid, ctx, e_t] -> logits straight into outputs[t+1]
        k_copy_cols<<<Bz, 128, 0, stream>>>(hid, Dz, cat, KCAT, Dz);
        k_copy_cols<<<Bz, 128, 0, stream>>>(ctx, C2, cat + Dz, KCAT, C2);
        k_copy_cols<<<Bz, 128, 0, stream>>>(e_t, Ez, cat + Dz + C2, KCAT, Ez);
        gemm_nt_f32_wmma<<<dim3((Vz + 31) / 32, 1), 128, 0, stream>>>(
            cat, KCAT, out_W, KCAT, out_b, out + (size_t)(t + 1) * Bz * Vz, Vz,
            Bz, Vz, KCAT, 0);
    }
}